// EdgeWeightedGCNBlock_38371237822899
// MI455X (gfx1250) — compile-verified
//
#include <hip/hip_runtime.h>
#include <math.h>

typedef __attribute__((ext_vector_type(2))) float v2f;
typedef __attribute__((ext_vector_type(8))) float v8f;

#define DIM 128
#define LN_EPS 1e-5f

// ---------------------------------------------------------------------------
// Zero the output accumulator (d_out doubles as agg) and the degree buffer.
// ---------------------------------------------------------------------------
__global__ void zero_fill_kernel(float* __restrict__ out, long n_out,
                                 float* __restrict__ deg, int n_deg) {
    long i = (long)blockIdx.x * blockDim.x + threadIdx.x;
    if (i < n_out) out[i] = 0.0f;
    if (i < n_deg) deg[i] = 0.0f;
}

// ---------------------------------------------------------------------------
// deg[col[e]] += ew[e]  (float atomics, L2-resident: deg is 400 KB)
// ---------------------------------------------------------------------------
__global__ void deg_kernel(const int* __restrict__ col,
                           const float* __restrict__ ew,
                           float* __restrict__ deg, int E) {
    int e = blockIdx.x * blockDim.x + threadIdx.x;
    if (e < E) atomicAdd(&deg[col[e]], ew[e]);
}

// ---------------------------------------------------------------------------
// dinv = deg > 0 ? rsqrt(deg) : 0   (in place)
// ---------------------------------------------------------------------------
__global__ void dinv_kernel(float* __restrict__ d, int N) {
    int i = blockIdx.x * blockDim.x + threadIdx.x;
    if (i < N) {
        float v = d[i];
        d[i] = (v > 0.0f) ? rsqrtf(v) : 0.0f;
    }
}

// ---------------------------------------------------------------------------
// h = x @ W using V_WMMA_F32_16X16X4_F32 (exact f32 path).
// 256 threads = 8 waves; each wave computes a 16x128 output tile.
//
// W staged TRANSPOSED in LDS with XOR swizzle:
//     wt[n][ c(k,n) ],  c(k,n) = k ^ (2*(n&15)) ^ ((k&2)<<4)
//  -> (k,k+1) pairs contiguous & 8B-aligned: single ds_load_b64 straight into
//     an even-aligned WMMA B operand; conflict-free across all 32 lanes.
// A slab (one x row per lane, 64 floats) preloaded into registers.
// K-loop unrolled by 4 only: keeps VGPRs < 256 (no s_set_vgpr_msb traffic,
// >=2 waves/SIMD) while the inner loop stays pure ds_load_b64 + v_wmma.
// ---------------------------------------------------------------------------
__global__ void __launch_bounds__(256)
gemm_wmma_kernel(const float* __restrict__ x, const float* __restrict__ W,
                 float* __restrict__ h, int n_tiles) {
    __shared__ float lds_wt[DIM * DIM];  // 64 KB, transposed + swizzled

    // Cooperative fill: global reads coalesced (consecutive threads -> n).
    for (int idx = threadIdx.x; idx < DIM * DIM; idx += 256) {
        int k = idx >> 7;
        int n = idx & 127;
        int c = k ^ ((n & 15) << 1) ^ ((k & 2) << 4);
        lds_wt[n * DIM + c] = W[idx];
    }
    __syncthreads();

    const int wave = threadIdx.x >> 5;
    const int lane = threadIdx.x & 31;
    const int half = lane >> 4;   // K-pair selector within A/B layout
    const int l16  = lane & 15;   // M (for A) / N (for B) within the 16-tile

    const int tile = blockIdx.x * 8 + wave;   // uniform per wave -> EXEC all-1s
    if (tile >= n_tiles) return;

    const long row0 = (long)tile * 16;
    const float* xrow = x + (row0 + l16) * DIM;

    // Preload entire A slab for this lane: A[i] = x[row][4i+2*half .. +1]
    v2f areg[32];
    #pragma unroll
    for (int i = 0; i < 32; ++i)
        areg[i] = *(const v2f*)(xrow + i * 4 + half * 2);   // 8B-aligned

    v8f acc[8];
    #pragma unroll
    for (int n = 0; n < 8; ++n) acc[n] = (v8f){};

    // Per-lane swizzle constant: col = k0 ^ cb   (ka = k0 + 2*half, k0 % 4 == 0)
    const int cb = (half << 1) ^ (l16 << 1) ^ (half << 5);
    const float* bbase = lds_wt + l16 * DIM;

    #pragma unroll 4
    for (int i = 0; i < 32; ++i) {
        const int col = (i * 4) ^ cb;
        const float* bp = bbase + col;      // + ntile*16*DIM folds into ds offset
        #pragma unroll
        for (int n = 0; n < 8; ++n) {
            v2f b = *(const v2f*)(bp + n * 16 * DIM);
            acc[n] = __builtin_amdgcn_wmma_f32_16x16x4_f32(
                false, areg[i], false, b, (short)0, acc[n], false, false);
        }
    }

    // D layout: VGPR r holds (M = half*8 + r, N = n*16 + l16)
    #pragma unroll
    for (int n = 0; n < 8; ++n) {
        #pragma unroll
        for (int r = 0; r < 8; ++r) {
            h[(row0 + half * 8 + r) * DIM + n * 16 + l16] = acc[n][r];
        }
    }
}

// ---------------------------------------------------------------------------
// Edge scatter: one wave32 per edge, 4 features per lane.
// agg[col] += h[row] * (dinv[row]*ew*dinv[col]) via global_atomic_add_f32.
// h and agg both fit in the 192 MB L2, so atomics stay on-chip.
// ---------------------------------------------------------------------------
__global__ void __launch_bounds__(256)
scatter_kernel(const int* __restrict__ rowi, const int* __restrict__ coli,
               const float* __restrict__ ew, const float* __restrict__ dinv,
               const float* __restrict__ h, float* __restrict__ agg, int E) {
    int e = (int)(((long)blockIdx.x * blockDim.x + threadIdx.x) >> 5);
    int lane = threadIdx.x & 31;
    if (e >= E) return;

    int r = rowi[e];
    int c = coli[e];
    float nrm = dinv[r] * ew[e] * dinv[c];

    float4 v = ((const float4*)(h + (long)r * DIM))[lane];
    float* dst = agg + (long)c * DIM + lane * 4;
    atomicAdd(dst + 0, v.x * nrm);
    atomicAdd(dst + 1, v.y * nrm);
    atomicAdd(dst + 2, v.z * nrm);
    atomicAdd(dst + 3, v.w * nrm);
}

// ---------------------------------------------------------------------------
// Epilogue: per node (one wave32 each): agg += b -> LayerNorm -> GELU(erf)
// -> + x residual, in place on d_out. Wave butterfly reductions (wave32).
// ---------------------------------------------------------------------------
__global__ void __launch_bounds__(256)
epilogue_kernel(const float* __restrict__ x, const float* __restrict__ b,
                const float* __restrict__ gamma, const float* __restrict__ beta,
                float* __restrict__ out, int N) {
    int n = (int)(((long)blockIdx.x * blockDim.x + threadIdx.x) >> 5);
    int lane = threadIdx.x & 31;
    if (n >= N) return;

    long base = (long)n * DIM + lane * 4;
    float4 a  = *(float4*)(out + base);
    float4 bb = *(const float4*)(b + lane * 4);
    a.x += bb.x; a.y += bb.y; a.z += bb.z; a.w += bb.w;

    float s = a.x + a.y + a.z + a.w;
    #pragma unroll
    for (int off = 16; off > 0; off >>= 1) s += __shfl_xor(s, off, 32);
    float mu = s * (1.0f / 128.0f);

    float dx = a.x - mu, dy = a.y - mu, dz = a.z - mu, dw = a.w - mu;
    float ss = dx * dx + dy * dy + dz * dz + dw * dw;
    #pragma unroll
    for (int off = 16; off > 0; off >>= 1) ss += __shfl_xor(ss, off, 32);
    float rstd = rsqrtf(ss * (1.0f / 128.0f) + LN_EPS);

    float4 g  = *(const float4*)(gamma + lane * 4);
    float4 bt = *(const float4*)(beta + lane * 4);
    float4 xr = *(const float4*)(x + base);

    const float inv_sqrt2 = 0.70710678118654752f;
    float o4[4];
    float d4[4]  = {dx, dy, dz, dw};
    float g4[4]  = {g.x, g.y, g.z, g.w};
    float bt4[4] = {bt.x, bt.y, bt.z, bt.w};
    float xr4[4] = {xr.x, xr.y, xr.z, xr.w};
    #pragma unroll
    for (int i = 0; i < 4; ++i) {
        float ln = d4[i] * rstd * g4[i] + bt4[i];
        float gelu = 0.5f * ln * (1.0f + erff(ln * inv_sqrt2));
        o4[i] = xr4[i] + gelu;
    }
    float4 res = {o4[0], o4[1], o4[2], o4[3]};
    *(float4*)(out + base) = res;
}

// ---------------------------------------------------------------------------
extern "C" void kernel_launch(void* const* d_in, const int* in_sizes, int n_in,
                              void* d_out, int out_size, void* d_ws, size_t ws_size,
                              hipStream_t stream) {
    const float* x     = (const float*)d_in[0];
    const int*   ei    = (const int*)d_in[1];
    const float* ew    = (const float*)d_in[2];
    const float* W     = (const float*)d_in[3];
    const float* bias  = (const float*)d_in[4];
    const float* gamma = (const float*)d_in[5];
    const float* beta  = (const float*)d_in[6];

    const int N = in_sizes[0] / DIM;
    const int E = in_sizes[2];
    const int* rowi = ei;       // edge_index[0]
    const int* coli = ei + E;   // edge_index[1]

    float* h    = (float*)d_ws;                 // N*DIM floats (51.2 MB)
    float* dinv = h + (size_t)N * DIM;          // N floats (reused: deg -> dinv)
    float* out  = (float*)d_out;                // doubles as agg accumulator

    const long tot = (long)N * DIM;

    // 1) zero agg accumulator (d_out) + degree buffer
    {
        long blocks = (tot + 255) / 256;
        zero_fill_kernel<<<(int)blocks, 256, 0, stream>>>(out, tot, dinv, N);
    }
    // 2) degree accumulation
    deg_kernel<<<(E + 255) / 256, 256, 0, stream>>>(coli, ew, dinv, E);
    // 3) dinv = rsqrt(deg) (in place)
    dinv_kernel<<<(N + 255) / 256, 256, 0, stream>>>(dinv, N);
    // 4) h = x @ W via WMMA f32 16x16x4
    {
        int n_tiles = N / 16;   // N = 100000 is divisible by 16
        gemm_wmma_kernel<<<(n_tiles + 7) / 8, 256, 0, stream>>>(x, W, h, n_tiles);
    }
    // 5) edge gather/scale/scatter-add into d_out
    {
        long threads = (long)E * 32;
        scatter_kernel<<<(int)((threads + 255) / 256), 256, 0, stream>>>(
            rowi, coli, ew, dinv, h, out, E);
    }
    // 6) bias + LayerNorm + GELU + residual, in place on d_out
    {
        long threads = (long)N * 32;
        epilogue_kernel<<<(int)((threads + 255) / 256), 256, 0, stream>>>(
            x, bias, gamma, beta, out, N);
    }
}